// End2EndRVTwoModels_90933047590946
// MI455X (gfx1250) — compile-verified
//
#include <hip/hip_runtime.h>

// ---------------------------------------------------------------------------
// Problem constants (shapes fixed by setup_inputs in the reference)
// ---------------------------------------------------------------------------
#define BATCH   4
#define N1      25200
#define NTOT    50400          // N1 + N2 anchors per batch
#define C1      85             // 4 box + 1 conf + 80 classes
#define C2      25             // 4 box + 1 conf + 20 classes
#define NL1     80
#define NL2     20
#define MAXOBJ  100
#define MAX_WH_F     7680.0f
#define SCORE_THR    0.25f
#define IOU_THR      0.45f

typedef float v2f  __attribute__((ext_vector_type(2)));
typedef float v8f  __attribute__((ext_vector_type(8)));
typedef unsigned int u32x4 __attribute__((ext_vector_type(4)));
typedef int  i32x4 __attribute__((ext_vector_type(4)));
typedef int  i32x8 __attribute__((ext_vector_type(8)));

// Toolchain discriminator: the 6-arg tensor_load_to_lds builtin ships with the
// toolchain that also provides the gfx1250 TDM header; ROCm 7.2 uses 5 args.
#if defined(__has_include)
#  if __has_include(<hip/amd_detail/amd_gfx1250_TDM.h>)
#    define TDM_SIX_ARG 1
#  else
#    define TDM_SIX_ARG 0
#  endif
#else
#  define TDM_SIX_ARG 0
#endif

// ---------------------------------------------------------------------------
// Kernel 1: boxes = x[:, :, 0:4] @ CONVERT via V_WMMA_F32_16X16X4_F32.
// One wave = 16 anchors. A (16x4 f32): lanes 0-15 supply K0/K1, lanes 16-31
// K2/K3. B = CONVERT zero-padded to 4x16, built from lane id (exact fp32).
// D: lane n<16 -> (M=m,N=n); lane n>=16 -> (M=m+8,N=n-16). Store N<4 only.
// boxes[] is flat-indexed by global anchor, so store addr = (base+M)*4+N
// (no divisions -- fixes the div-by-50400 chains seen in round 1).
// ---------------------------------------------------------------------------
__global__ __launch_bounds__(256) void box_wmma_kernel(
    const float* __restrict__ x1, const float* __restrict__ x2,
    float* __restrict__ boxes)
{
  const int lane = threadIdx.x & 31;
  const int wave = (blockIdx.x * blockDim.x + threadIdx.x) >> 5;
  const int base = wave * 16;                 // 16 anchors per wave (exact cover)
  const int m    = lane & 15;
  const int aIdx = base + m;                  // flat anchor loaded by this lane
  const int b    = aIdx / NTOT;
  const int n    = aIdx % NTOT;
  const int c0   = (lane < 16) ? 0 : 2;       // channels this lane supplies (K rows)

  const float* row = (n < N1)
      ? (x1 + ((size_t)b * N1 + n) * C1)
      : (x2 + ((size_t)b * N1 + (n - N1)) * C2);

#if defined(__AMDGCN__) && __has_builtin(__builtin_amdgcn_wmma_f32_16x16x4_f32)
  v2f A;
  A.x = row[c0];
  A.y = row[c0 + 1];

  // CONVERT rows: k0={1,0,1,0} k1={0,1,0,1} k2={-.5,0,.5,0} k3={0,-.5,0,.5}
  const int ncol = lane & 15;                 // B column N
  float b0, b1;
  if (lane < 16) {                            // rows K=0 (b0), K=1 (b1)
    b0 = (ncol == 0 || ncol == 2) ? 1.0f : 0.0f;
    b1 = (ncol == 1 || ncol == 3) ? 1.0f : 0.0f;
  } else {                                    // rows K=2 (b0), K=3 (b1)
    b0 = (ncol == 0) ? -0.5f : ((ncol == 2) ? 0.5f : 0.0f);
    b1 = (ncol == 1) ? -0.5f : ((ncol == 3) ? 0.5f : 0.0f);
  }
  v2f Bm; Bm.x = b0; Bm.y = b1;

  v8f Cz = {};
  v8f D = __builtin_amdgcn_wmma_f32_16x16x4_f32(
      /*neg_a=*/false, A, /*neg_b=*/false, Bm,
      /*c_mod=*/(short)0, Cz, /*reuse_a=*/false, /*reuse_b=*/false);

  if (ncol < 4) {                             // only N=0..3 carry real outputs
    const int Mbase = (lane < 16) ? 0 : 8;
    float* o = boxes + (size_t)(base + Mbase) * 4 + ncol;
#pragma unroll
    for (int r = 0; r < 8; ++r)
      o[(size_t)r * 4] = D[r];
  }
#else
  // Scalar fallback (also what the host pass parses): lanes 0-15, one anchor each.
  if (lane < 16) {
    const float cx = row[0], cy = row[1], w = row[2], h = row[3];
    float* o = boxes + (size_t)aIdx * 4;
    o[0] = cx - 0.5f * w;
    o[1] = cy - 0.5f * h;
    o[2] = cx + 0.5f * w;
    o[3] = cy + 0.5f * h;
  }
#endif
}

// ---------------------------------------------------------------------------
// Kernel 2: per-anchor max_score / cat_id (scores = class * conf), streaming,
// bandwidth-bound (~44 MB once => ~1.9 us at 23.3 TB/s). Reproduces the
// zero-padding semantics: x1 -> real ids 0..79 (zeros 80..99), x2 -> zeros
// 0..79 (real ids 80..99); jnp.argmax first-max tie rules included.
// ---------------------------------------------------------------------------
__global__ __launch_bounds__(256) void score_kernel(
    const float* __restrict__ x1, const float* __restrict__ x2,
    float* __restrict__ score, int* __restrict__ cat)
{
  const int tid = blockIdx.x * blockDim.x + threadIdx.x;
  if (tid >= BATCH * NTOT) return;
  const int b = tid / NTOT, n = tid % NTOT;

  float best; int bi;
  if (n < N1) {
    const float* row = x1 + ((size_t)b * N1 + n) * C1;
    __builtin_prefetch(row + C1, 0, 3);       // -> global_prefetch_b8
    const float conf = row[4];
    best = row[5] * conf; bi = 0;
#pragma unroll 4
    for (int c = 1; c < NL1; ++c) {
      const float v = row[5 + c] * conf;
      if (v > best) { best = v; bi = c; }
    }
    if (best < 0.0f) { best = 0.0f; bi = NL1; }   // padding ids 80.. win only then
  } else {
    const float* row = x2 + ((size_t)b * N1 + (n - N1)) * C2;
    __builtin_prefetch(row + C2, 0, 3);
    const float conf = row[4];
    best = row[5] * conf; bi = NL1;
#pragma unroll 4
    for (int c = 1; c < NL2; ++c) {
      const float v = row[5 + c] * conf;
      if (v > best) { best = v; bi = NL1 + c; }
    }
    if (best <= 0.0f) { best = 0.0f; bi = 0; }    // zeros at ids 0..79 come first
  }
  score[tid] = best;
  cat[tid]   = bi;
}

// ---------------------------------------------------------------------------
// Kernel 3: sequential NMS (100 steps) + gather. One 1024-thread block per
// batch. Live-score array (dead = -1.0, identical to ref's where(valid,s,-1))
// lives in 201.6 KB of dynamic LDS, initialized by one TDM tensor_load_to_lds
// per workgroup (D# built per ISA 8.3/8.4) + s_wait_tensorcnt. Argmax via LDS
// tree reduction with first-index tie-break; boxes/cat stream from L2;
// nmsbox = box + cat*7680 recomputed in fp32 (exact product, same rounding).
// ---------------------------------------------------------------------------
__global__ __launch_bounds__(1024) void nms_kernel(
    const float* __restrict__ boxes, const float* __restrict__ score,
    const int* __restrict__ cat, float* __restrict__ out)
{
  extern __shared__ float sScore[];           // NTOT floats (201.6 KB dynamic LDS)
  __shared__ float rval[1024];
  __shared__ int   ridx[1024];
  __shared__ float bbx[4];
  __shared__ int   curI;
  __shared__ int   selIdx[MAXOBJ];

  const int b  = blockIdx.x;
  const int t  = threadIdx.x;
  const int NT = 1024;

  const float* boxesB = boxes + (size_t)b * NTOT * 4;
  const float* scoreB = score + (size_t)b * NTOT;
  const int*   catB   = cat   + (size_t)b * NTOT;

#if defined(__AMDGCN__) && __has_builtin(__builtin_amdgcn_tensor_load_to_lds) && \
    __has_builtin(__builtin_amdgcn_s_wait_tensorcnt)
  if (t < 32) {                               // wave 0 issues the TDM (EXEC ignored)
    const unsigned long long ga = (unsigned long long)(const void*)scoreB;
    const unsigned ldsOff = (unsigned)(unsigned long long)(void*)sScore; // addr[31:0] = LDS offset

    u32x4 g0;
    g0[0] = 1u;                                               // count=1, user desc
    g0[1] = ldsOff;                                           // lds_addr (bytes)
    g0[2] = (unsigned)(ga & 0xFFFFFFFFull);                   // global_addr[31:0]
    g0[3] = (unsigned)((ga >> 32) & 0x01FFFFFFull)            // global_addr[56:32]
          | (2u << 30);                                       // type = 2 ("image")

    // 2-D tensor: dim0 = tile0 = 50400 f32, dim1 = tile1 = 1, stride0 = 50400
    i32x8 g1;
    g1[0] = 0x00020000;                                       // data_size = 4 B
    g1[1] = (int)(((unsigned)NTOT & 0xFFFFu) << 16);          // tensor_dim0[15:0]
    g1[2] = (int)(((unsigned)NTOT >> 16) | (1u << 16));       // dim0[31:16] | tensor_dim1[15:0]=1
    g1[3] = (int)(((unsigned)NTOT & 0xFFFFu) << 16);          // dim1[31:16]=0 | tile_dim0
    g1[4] = 1;                                                // tile_dim1=1, tile_dim2=0
    g1[5] = NTOT;                                             // tensor_dim0_stride[31:0]
    g1[6] = (int)(((unsigned)NTOT & 0xFFFFu) << 16);          // stride0[47:32]=0 | stride1[15:0]
    g1[7] = 0;                                                // stride1[47:16]

    i32x4 gz = {0, 0, 0, 0};                                  // groups 2/3 unused (<=2D)
#if TDM_SIX_ARG
    i32x8 gz8 = {0, 0, 0, 0, 0, 0, 0, 0};
    __builtin_amdgcn_tensor_load_to_lds(g0, g1, gz, gz, gz8, 0);
#else
    __builtin_amdgcn_tensor_load_to_lds(g0, g1, gz, gz, 0);
#endif
    __builtin_amdgcn_s_wait_tensorcnt(0);
  }
  __syncthreads();
  // valid0 = score >= 0.25 ; dead entries become exactly -1.0 like the ref
  for (int j = t; j < NTOT; j += NT) {
    const float v = sScore[j];
    sScore[j] = (v >= SCORE_THR) ? v : -1.0f;
  }
#else
  for (int j = t; j < NTOT; j += NT) {
    const float v = scoreB[j];
    sScore[j] = (v >= SCORE_THR) ? v : -1.0f;
  }
#endif
  __syncthreads();

  for (int k = 0; k < MAXOBJ; ++k) {
    // ---- argmax over live scores (first max wins) ----
    float best = -2.0f; int bidx = 0x7fffffff;
    for (int j = t; j < NTOT; j += NT) {
      const float s = sScore[j];
      if (s > best) { best = s; bidx = j; }
    }
    rval[t] = best; ridx[t] = bidx;
    __syncthreads();
    for (int s2 = NT >> 1; s2 > 0; s2 >>= 1) {
      if (t < s2) {
        const float v1 = rval[t], v2 = rval[t + s2];
        const int   i1 = ridx[t], i2 = ridx[t + s2];
        if (v2 > v1 || (v2 == v1 && i2 < i1)) { rval[t] = v2; ridx[t] = i2; }
      }
      __syncthreads();
    }
    if (t == 0) {
      const int   i = ridx[0];
      const float v = rval[0];
      selIdx[k] = (v >= SCORE_THR) ? i : -1;
      curI = i;
      const float off = (float)catB[i] * MAX_WH_F;  // exact product; same rounding as ref
      bbx[0] = boxesB[i * 4 + 0] + off;
      bbx[1] = boxesB[i * 4 + 1] + off;
      bbx[2] = boxesB[i * 4 + 2] + off;
      bbx[3] = boxesB[i * 4 + 3] + off;
    }
    __syncthreads();

    // ---- suppression: live &= (iou <= 0.45); live[i] = dead ----
    const float p0 = bbx[0], p1 = bbx[1], p2 = bbx[2], p3 = bbx[3];
    const int   ci = curI;
    const float areaA = (p2 - p0) * (p3 - p1);
    for (int j = t; j < NTOT; j += NT) {
      if (j == ci) { sScore[j] = -1.0f; continue; }
      if (sScore[j] < 0.0f) continue;               // already dead
      const float off = (float)catB[j] * MAX_WH_F;
      const float q0 = boxesB[j * 4 + 0] + off;
      const float q1 = boxesB[j * 4 + 1] + off;
      const float q2 = boxesB[j * 4 + 2] + off;
      const float q3 = boxesB[j * 4 + 3] + off;
      const float ix1 = fmaxf(p0, q0), iy1 = fmaxf(p1, q1);
      const float ix2 = fminf(p2, q2), iy2 = fminf(p3, q3);
      const float inter = fmaxf(ix2 - ix1, 0.0f) * fmaxf(iy2 - iy1, 0.0f);
      const float areaB = (q2 - q0) * (q3 - q1);
      const float iou = inter / (areaA + areaB - inter + 1e-9f);
      if (!(iou <= IOU_THR)) sScore[j] = -1.0f;     // NaN also kills, like the ref
    }
    __syncthreads();
  }

  // ---- gather output rows: [X, box*4, cat, score] ----
  if (t < MAXOBJ) {
    const int  i  = selIdx[t];
    const bool m  = (i >= 0);
    const int  si = m ? i : 0;
    const float mf = m ? 1.0f : 0.0f;
    float* o = out + ((size_t)b * MAXOBJ + t) * 7;
    o[0] = m ? (float)b : -1.0f;
    o[1] = boxesB[si * 4 + 0] * mf;
    o[2] = boxesB[si * 4 + 1] * mf;
    o[3] = boxesB[si * 4 + 2] * mf;
    o[4] = boxesB[si * 4 + 3] * mf;
    o[5] = (float)catB[si] * mf;
    o[6] = scoreB[si] * mf;                          // original score, unmodified
  }
}

// ---------------------------------------------------------------------------
extern "C" void kernel_launch(void* const* d_in, const int* in_sizes, int n_in,
                              void* d_out, int out_size, void* d_ws, size_t ws_size,
                              hipStream_t stream)
{
  const float* x1 = (const float*)d_in[0];
  const float* x2 = (const float*)d_in[1];
  float* out = (float*)d_out;

  // workspace: boxes (B*N*4 f32) | score (B*N f32) | cat (B*N i32)  ~4.84 MB
  float* boxes = (float*)d_ws;
  float* score = boxes + (size_t)BATCH * NTOT * 4;
  int*   cat   = (int*)(score + (size_t)BATCH * NTOT);

  // Kernel 1: (B*N)/16 = 12600 waves, 8 waves per 256-thread block -> 1575 blocks
  const int blocks1 = (BATCH * NTOT) / (16 * 8);
  hipLaunchKernelGGL(box_wmma_kernel, dim3(blocks1), dim3(256), 0, stream,
                     x1, x2, boxes);

  const int blocks2 = (BATCH * NTOT + 255) / 256;
  hipLaunchKernelGGL(score_kernel, dim3(blocks2), dim3(256), 0, stream,
                     x1, x2, score, cat);

  // Kernel 3: one workgroup per batch; 201.6 KB dynamic LDS (< 320 KB/WGP)
  hipLaunchKernelGGL(nms_kernel, dim3(BATCH), dim3(1024),
                     NTOT * sizeof(float), stream,
                     boxes, score, cat, out);
}